// SummationMPNN_39444979646941
// MI455X (gfx1250) — compile-verified
//
#include <hip/hip_runtime.h>
#include <hip/hip_bf16.h>
#include <math.h>

// ---------------------------------------------------------------------------
// SummationMPNN on MI455X (gfx1250, wave32).
// GEMMs: V_WMMA_F32_16X16X4_F32 (fp32; problem is bandwidth-bound, precision
// is free). Bulk copies: async Global->LDS B128 (GLOBAL_LOAD_ASYNC_TO_LDS_B128,
// ASYNCcnt-tracked) when the toolchain exposes the builtin, else float4.
// ---------------------------------------------------------------------------

#define Bsz     8
#define Nn      256
#define Fn      64      // F_NODE
#define Fe      16      // F_EDGE
#define Mm      64      // M_MSG
#define G3      192     // 3*F_NODE
#define OUTD    128

typedef __attribute__((ext_vector_type(2))) float v2f;
typedef __attribute__((ext_vector_type(8))) float v8f;
typedef int v4i_ __attribute__((vector_size(16)));   // matches builtin param

#define AS1 __attribute__((address_space(1)))
#define AS3 __attribute__((address_space(3)))

#if defined(__gfx1250__) && __has_builtin(__builtin_amdgcn_global_load_async_to_lds_b128)
#define HAVE_ASYNC_LDS 1
#endif

// Copy 16 bytes global -> LDS (per-lane). Async path uses ASYNCcnt.
__device__ __forceinline__ void copy16_g2l(const float* __restrict__ gsrc,
                                           float* __restrict__ ldst) {
#ifdef HAVE_ASYNC_LDS
    __builtin_amdgcn_global_load_async_to_lds_b128(
        (AS1 v4i_*)(v4i_*)(float*)gsrc,
        (AS3 v4i_*)(v4i_*)ldst,
        /*offset=*/0, /*cpol=*/0);
#else
    *(float4*)ldst = *(const float4*)gsrc;
#endif
}

__device__ __forceinline__ void copy_wait() {
#ifdef HAVE_ASYNC_LDS
#if __has_builtin(__builtin_amdgcn_s_wait_asynccnt)
    __builtin_amdgcn_s_wait_asynccnt(0);
#else
    asm volatile("s_wait_asynccnt 0x0" ::: "memory");
#endif
#endif
}

// D = A(16x4) * B(4x16) + C  (fp32 WMMA, wave32)
__device__ __forceinline__ v8f wmma4(v2f a, v2f b, v8f c) {
    return __builtin_amdgcn_wmma_f32_16x16x4_f32(
        /*neg_a=*/false, a, /*neg_b=*/false, b,
        /*c_mod=*/(short)0, c, /*reuse_a=*/false, /*reuse_b=*/false);
}

__device__ __forceinline__ float sigmf(float x) {
    return 1.0f / (1.0f + __expf(-x));
}

// ---------------------------------------------------------------------------
// init: h <- nodes ; nodemask[b,i] = any_j ( sum_k edges[b,i,j,k] != 0 )
// grid: B*N blocks of 256 threads (thread t handles neighbour j=t)
// ---------------------------------------------------------------------------
__global__ void mpnn_init(const float* __restrict__ nodes,
                          const float* __restrict__ edges,
                          float* __restrict__ hbuf,
                          float* __restrict__ nodemask) {
    int bi = blockIdx.x;               // flat (b*N + i)
    int t  = threadIdx.x;
    if (t < Fn) hbuf[(size_t)bi * Fn + t] = nodes[(size_t)bi * Fn + t];
    const float4* ep4 = (const float4*)(edges + ((size_t)bi * Nn + t) * Fe);
    float s = 0.f;
    #pragma unroll
    for (int r = 0; r < 4; ++r) {      // 16 edge feats = 4 x B128
        float4 v = ep4[r];
        s += v.x + v.y + v.z + v.w;
    }
    __shared__ int anyflag;
    if (t == 0) anyflag = 0;
    __syncthreads();
    if (s != 0.f) atomicOr(&anyflag, 1);
    __syncthreads();
    if (t == 0) nodemask[bi] = anyflag ? 1.0f : 0.0f;
}

// ---------------------------------------------------------------------------
// Hm[b,j,:] = h[b,j,:] @ W_msg[0:64,:] + b_msg      (j-only part of message MLP)
// grid: (B*N/16) blocks, 128 threads (4 waves, one 16-col N-tile each), K=64
// ---------------------------------------------------------------------------
__global__ void mpnn_hm(const float* __restrict__ hbuf,
                        const float* __restrict__ W_msg,
                        const float* __restrict__ b_msg,
                        float* __restrict__ Hm) {
    int r0   = blockIdx.x * 16;
    int lane = threadIdx.x & 31;
    int w    = threadIdx.x >> 5;       // 0..3
    __shared__ alignas(16) float aT[16 * Fn];
    for (int q = threadIdx.x; q < 256; q += 128)          // 256 x B128
        copy16_g2l(hbuf + (size_t)r0 * Fn + q * 4, aT + q * 4);
    copy_wait();
    __syncthreads();
    int n = lane & 15, h2 = lane >> 4;
    int c = w * 16 + n;
    v8f acc = {};
    #pragma unroll
    for (int kb = 0; kb < 16; ++kb) {
        int k = kb * 4 + h2 * 2;
        v2f a;  a.x  = aT[(lane & 15) * Fn + k];
                a.y  = aT[(lane & 15) * Fn + k + 1];
        v2f bb; bb.x = W_msg[k * Mm + c];
                bb.y = W_msg[(k + 1) * Mm + c];
        acc = wmma4(a, bb, acc);
    }
    float bias = b_msg[c];
    #pragma unroll
    for (int r = 0; r < 8; ++r) {
        int m = r + 8 * h2;
        Hm[(size_t)(r0 + m) * Mm + c] = acc[r] + bias;
    }
}

// ---------------------------------------------------------------------------
// msg[b,i,:] = sum_j mask[b,i,j] * relu( edges[b,i,j,:]@W_msg[64:80,:] + Hm[b,j,:] )
// grid: B*N blocks (one destination node each), 128 threads (4 waves).
// Each wave owns a 16-col slice of the 64-wide message; loop over 16 j-tiles.
// Per tile the edges block (1KB) and Hm block (4KB) are contiguous -> async
// B128 Global->LDS copies.
// ---------------------------------------------------------------------------
__global__ void mpnn_msg(const float* __restrict__ edges,
                         const float* __restrict__ Hm,
                         const float* __restrict__ W_msg,
                         float* __restrict__ msg) {
    int bi   = blockIdx.x;             // b*N + i
    int b    = bi >> 8;                // N = 256
    int tid  = threadIdx.x;
    int lane = tid & 31;
    int w    = tid >> 5;               // 0..3
    __shared__ alignas(16) float eT[16 * Fe];   // 16 j-rows x 16 edge feats
    __shared__ alignas(16) float hmT[16 * Mm];  // 16 j-rows x 64
    __shared__ float maskT[16];
    int n = lane & 15, h2 = lane >> 4;
    int c = w * 16 + n;
    const float* eBase  = edges + (size_t)bi * Nn * Fe;   // 256x16, contiguous
    const float* hmBase = Hm + (size_t)b * Nn * Mm;       // 256x64, contiguous
    float accLane = 0.f;
    for (int jt = 0; jt < 16; ++jt) {
        __syncthreads();               // previous tile fully consumed
        // edges tile: 1KB contiguous = 64 x B128 (waves 0,1)
        if (tid < 64)
            copy16_g2l(eBase + (size_t)jt * 256 + tid * 4, eT + tid * 4);
        // Hm tile: 4KB contiguous = 256 x B128 (2 per thread)
        for (int q = tid; q < 256; q += 128)
            copy16_g2l(hmBase + (size_t)jt * 1024 + q * 4, hmT + q * 4);
        // prefetch next tile's edge block while this tile computes
        if (jt < 15)
            __builtin_prefetch(eBase + (size_t)(jt + 1) * 256 + tid * 2, 0, 1);
        copy_wait();
        __syncthreads();               // all copies landed
        if (tid < 16) {
            float s = 0.f;
            #pragma unroll
            for (int k = 0; k < Fe; ++k) s += eT[tid * Fe + k];
            maskT[tid] = (s != 0.f) ? 1.0f : 0.0f;
        }
        __syncthreads();
        v8f acc = {};
        #pragma unroll
        for (int kb = 0; kb < 4; ++kb) {          // K = 16
            int k = kb * 4 + h2 * 2;
            v2f a;  a.x  = eT[(lane & 15) * Fe + k];
                    a.y  = eT[(lane & 15) * Fe + k + 1];
            v2f bb; bb.x = W_msg[(Fn + k) * Mm + c];
                    bb.y = W_msg[(Fn + k + 1) * Mm + c];
            acc = wmma4(a, bb, acc);
        }
        #pragma unroll
        for (int r = 0; r < 8; ++r) {
            int m = r + 8 * h2;                   // j within tile
            float t = acc[r] + hmT[m * Mm + c];
            t = t > 0.f ? t : 0.f;
            accLane += maskT[m] * t;
        }
    }
    // combine lane halves (same n, disjoint m-sets)
    float other = __shfl_xor(accLane, 16, 32);
    if (lane < 16) msg[(size_t)bi * Mm + c] = accLane + other;
}

// ---------------------------------------------------------------------------
// GRU: gi = msg@W_i + b_i ; gh = h@W_h + b_h ; gated update, mask-guarded.
// grid: (B*N/16) blocks, 256 threads (8 waves). 24 WMMA tile-jobs (2 GEMMs x 12
// N-tiles of 16, K=64), 3 jobs/wave, results staged in LDS for elementwise.
// ---------------------------------------------------------------------------
__global__ void mpnn_gru(const float* __restrict__ msg,
                         const float* __restrict__ W_i,
                         const float* __restrict__ W_h,
                         const float* __restrict__ b_i,
                         const float* __restrict__ b_h,
                         const float* __restrict__ nodemask,
                         float* __restrict__ hbuf) {
    int r0   = blockIdx.x * 16;
    int tid  = threadIdx.x;
    int lane = tid & 31;
    int w    = tid >> 5;               // 0..7
    __shared__ alignas(16) float aM[16 * Fn], aH[16 * Fn];
    __shared__ alignas(16) float gi[16 * G3], gh[16 * G3];
    // 256 x B128 each, one per thread
    copy16_g2l(msg  + (size_t)r0 * Fn + tid * 4, aM + tid * 4);
    copy16_g2l(hbuf + (size_t)r0 * Fn + tid * 4, aH + tid * 4);
    copy_wait();
    __syncthreads();
    int n = lane & 15, h2 = lane >> 4;
    for (int job = w; job < 24; job += 8) {       // uniform: 3 per wave
        const float* A  = (job < 12) ? aM : aH;
        const float* Wt = (job < 12) ? W_i : W_h;
        float* Dst      = (job < 12) ? gi : gh;
        int nt = (job < 12) ? job : job - 12;
        int c  = nt * 16 + n;
        v8f acc = {};
        #pragma unroll
        for (int kb = 0; kb < 16; ++kb) {         // K = 64
            int k = kb * 4 + h2 * 2;
            v2f a;  a.x  = A[(lane & 15) * Fn + k];
                    a.y  = A[(lane & 15) * Fn + k + 1];
            v2f bb; bb.x = Wt[k * G3 + c];
                    bb.y = Wt[(k + 1) * G3 + c];
            acc = wmma4(a, bb, acc);
        }
        #pragma unroll
        for (int r = 0; r < 8; ++r)
            Dst[(r + 8 * h2) * G3 + c] = acc[r];
    }
    __syncthreads();
    for (int e = tid; e < 16 * Fn; e += 256) {
        int row = e >> 6, cc = e & 63;
        float ir  = gi[row * G3 + cc]       + b_i[cc];
        float iz  = gi[row * G3 + 64 + cc]  + b_i[64 + cc];
        float in_ = gi[row * G3 + 128 + cc] + b_i[128 + cc];
        float hr  = gh[row * G3 + cc]       + b_h[cc];
        float hz  = gh[row * G3 + 64 + cc]  + b_h[64 + cc];
        float hn  = gh[row * G3 + 128 + cc] + b_h[128 + cc];
        float rr  = sigmf(ir + hr);
        float zz  = sigmf(iz + hz);
        float ng  = tanhf(in_ + rr * hn);
        float hp  = aH[row * Fn + cc];
        float hnw = (1.0f - zz) * ng + zz * hp;
        float mf  = nodemask[r0 + row];
        hbuf[(size_t)(r0 + row) * Fn + cc] = (mf != 0.f) ? hnw : hp;
    }
}

// ---------------------------------------------------------------------------
// Readout: gate = sigmoid([h|x]@W_g + b_g) (K=128), emb = h@W_e + b_e (K=64),
// partial[b,itile,:] = sum over the 16 rows of nodemask*gate*emb.
// grid: B*16 blocks, 256 threads (8 waves, one 16-col tile each of 128 cols)
// ---------------------------------------------------------------------------
__global__ void mpnn_readout(const float* __restrict__ hbuf,
                             const float* __restrict__ nodes,
                             const float* __restrict__ W_g,
                             const float* __restrict__ b_g,
                             const float* __restrict__ W_e,
                             const float* __restrict__ b_e,
                             const float* __restrict__ nodemask,
                             float* __restrict__ partial) {
    int blk = blockIdx.x;              // 0..127
    int b   = blk >> 4;
    int i0  = (blk & 15) * 16;
    int r0  = b * Nn + i0;             // flat row into (B*N, F)
    int tid = threadIdx.x, lane = tid & 31, w = tid >> 5;
    __shared__ alignas(16) float aT[16 * 128];   // [h | x]
    __shared__ alignas(16) float gT[16 * OUTD], eTm[16 * OUTD];
    __shared__ float red[256];
    {   // each thread: one B128 of h, one B128 of x (16 rows x 16 float4 each)
        int row = tid >> 4, c4 = tid & 15;
        copy16_g2l(hbuf  + (size_t)(r0 + row) * Fn + c4 * 4,
                   aT + row * 128 + c4 * 4);
        copy16_g2l(nodes + (size_t)(r0 + row) * Fn + c4 * 4,
                   aT + row * 128 + 64 + c4 * 4);
        copy_wait();
    }
    __syncthreads();
    int n = lane & 15, h2 = lane >> 4;
    int c = w * 16 + n;
    v8f acc = {};
    #pragma unroll
    for (int kb = 0; kb < 32; ++kb) {             // gate: K = 128
        int k = kb * 4 + h2 * 2;
        v2f a;  a.x  = aT[(lane & 15) * 128 + k];
                a.y  = aT[(lane & 15) * 128 + k + 1];
        v2f bb; bb.x = W_g[k * OUTD + c];
                bb.y = W_g[(k + 1) * OUTD + c];
        acc = wmma4(a, bb, acc);
    }
    #pragma unroll
    for (int r = 0; r < 8; ++r) gT[(r + 8 * h2) * OUTD + c] = acc[r];
    v8f acc2 = {};
    #pragma unroll
    for (int kb = 0; kb < 16; ++kb) {             // emb: K = 64 (h part of aT)
        int k = kb * 4 + h2 * 2;
        v2f a;  a.x  = aT[(lane & 15) * 128 + k];
                a.y  = aT[(lane & 15) * 128 + k + 1];
        v2f bb; bb.x = W_e[k * OUTD + c];
                bb.y = W_e[(k + 1) * OUTD + c];
        acc2 = wmma4(a, bb, acc2);
    }
    #pragma unroll
    for (int r = 0; r < 8; ++r) eTm[(r + 8 * h2) * OUTD + c] = acc2[r];
    __syncthreads();
    int col = tid & 127, half = tid >> 7;
    float bg = b_g[col], be = b_e[col];
    float s = 0.f;
    #pragma unroll
    for (int rr = 0; rr < 8; ++rr) {
        int row = half * 8 + rr;
        float g  = sigmf(gT[row * OUTD + col] + bg);
        float em = eTm[row * OUTD + col] + be;
        s += nodemask[r0 + row] * g * em;
    }
    red[tid] = s;
    __syncthreads();
    if (tid < 128)
        partial[(size_t)blk * OUTD + col] = red[tid] + red[tid + 128];
}

// Deterministic final reduction over the 16 i-tiles per batch.
__global__ void mpnn_reduce(const float* __restrict__ partial,
                            float* __restrict__ out) {
    int b = blockIdx.x;                // 0..7
    int c = threadIdx.x;               // 0..127
    float s = 0.f;
    #pragma unroll
    for (int t = 0; t < 16; ++t)
        s += partial[((size_t)b * 16 + t) * OUTD + c];
    out[(size_t)b * OUTD + c] = s;
}

// ---------------------------------------------------------------------------
extern "C" void kernel_launch(void* const* d_in, const int* in_sizes, int n_in,
                              void* d_out, int out_size, void* d_ws, size_t ws_size,
                              hipStream_t stream) {
    const float* nodes = (const float*)d_in[0];
    const float* edges = (const float*)d_in[1];
    const float* W_msg = (const float*)d_in[2];
    const float* b_msg = (const float*)d_in[3];
    const float* W_i   = (const float*)d_in[4];
    const float* W_h   = (const float*)d_in[5];
    const float* b_i   = (const float*)d_in[6];
    const float* b_h   = (const float*)d_in[7];
    const float* W_g   = (const float*)d_in[8];
    const float* b_g   = (const float*)d_in[9];
    const float* W_e   = (const float*)d_in[10];
    const float* b_e   = (const float*)d_in[11];
    float* out = (float*)d_out;

    float* ws       = (float*)d_ws;
    float* hbuf     = ws;                         // B*N*F   = 131072
    float* Hm       = ws + 131072;                // B*N*M   = 131072
    float* msg      = ws + 262144;                // B*N*M   = 131072
    float* nodemask = ws + 393216;                // B*N     = 2048
    float* partial  = ws + 395264;                // 128*128 = 16384

    mpnn_init<<<Bsz * Nn, 256, 0, stream>>>(nodes, edges, hbuf, nodemask);
    for (int p = 0; p < 3; ++p) {
        mpnn_hm <<<Bsz * Nn / 16, 128, 0, stream>>>(hbuf, W_msg, b_msg, Hm);
        mpnn_msg<<<Bsz * Nn,      128, 0, stream>>>(edges, Hm, W_msg, msg);
        mpnn_gru<<<Bsz * Nn / 16, 256, 0, stream>>>(msg, W_i, W_h, b_i, b_h,
                                                    nodemask, hbuf);
    }
    mpnn_readout<<<Bsz * 16, 256, 0, stream>>>(hbuf, nodes, W_g, b_g, W_e, b_e,
                                               nodemask, partial);
    mpnn_reduce<<<Bsz, OUTD, 0, stream>>>(partial, out);
}